// CenterNet_80187039416841
// MI455X (gfx1250) — compile-verified
//
#include <hip/hip_runtime.h>
#include <stdint.h>

#define NB 8
#define NC 32
#define NH 512
#define NW 512
#define RADIUS 64
#define CROP 128            // 2*RADIUS
#define R_EXPAND 8
#define R_CLAMP 254         // R_MAX-1 = (H/2-1)-1

#define NCHUNK 64           // partial-argmax blocks per batch
#define CHUNK  (NH * NW / NCHUNK)   // 4096 elements per chunk

typedef unsigned int  u32;
typedef unsigned long long u64;
typedef u32 u32x4 __attribute__((ext_vector_type(4)));
typedef int i32x8 __attribute__((ext_vector_type(8)));
typedef int i32x4 __attribute__((ext_vector_type(4)));

#if __has_builtin(__builtin_amdgcn_tensor_load_to_lds)
#define USE_TDM 1
#elif __has_builtin(__builtin_amdgcn_global_load_async_to_lds_b128)
#define USE_ASYNC_LDS 1
#endif

// ws layout (ints):
//   [0 .. 63]                    params: 8 ints per batch (cx,cy,top,left,rad)
//   [64 .. 64+NB*NCHUNK)         partial vals (float bits)
//   [64+NB*NCHUNK .. +NB*NCHUNK) partial indices
#define WS_PARAM 0
#define WS_PVAL  64
#define WS_PIDX  (64 + NB * NCHUNK)

// ---------------------------------------------------------------------------
// Kernel 1a: partial fused 3x3-NMS + argmax over one 4096-element chunk.
// grid = NB*NCHUNK blocks x 256 threads -> enough WGPs to be HBM-limited.
// ---------------------------------------------------------------------------
__global__ __launch_bounds__(256) void nms_partial_kernel(
    const float* __restrict__ heatmap,   // [B,2,H,W]
    int* __restrict__ ws)
{
    const int b     = blockIdx.x / NCHUNK;
    const int chunk = blockIdx.x % NCHUNK;
    const int tid   = threadIdx.x;
    const float* __restrict__ heat = heatmap + ((size_t)b * 2 + 1) * (size_t)(NH * NW);
    const int base = chunk * CHUNK;

    float bestV = -1.0f;
    int   bestI = 0x7FFFFFFF;
    #pragma unroll 4
    for (int q = tid; q < CHUNK; q += 256) {
        const int p = base + q;
        const int i = p >> 9;          // row
        const int j = p & (NW - 1);    // col
        const float v = heat[p];
        // 3x3 window max, clamped (== reduce_window with -inf padding)
        const int i0 = (i > 0) ? i - 1 : i, i1 = (i < NH - 1) ? i + 1 : i;
        const int j0 = (j > 0) ? j - 1 : j, j1 = (j < NW - 1) ? j + 1 : j;
        float m = v;
        for (int ii = i0; ii <= i1; ++ii)
            for (int jj = j0; jj <= j1; ++jj)
                m = fmaxf(m, heat[ii * NW + jj]);
        const float cand = (m == v) ? v : 0.0f;  // NMS-suppressed value
        if (cand > bestV || (cand == bestV && p < bestI)) { bestV = cand; bestI = p; }
    }

    __shared__ float sV[256];
    __shared__ int   sI[256];
    sV[tid] = bestV; sI[tid] = bestI;
    __syncthreads();
    for (int s = 128; s > 0; s >>= 1) {
        if (tid < s) {
            const float ov = sV[tid + s]; const int oi = sI[tid + s];
            if (ov > sV[tid] || (ov == sV[tid] && oi < sI[tid])) { sV[tid] = ov; sI[tid] = oi; }
        }
        __syncthreads();
    }
    if (tid == 0) {
        ((float*)ws)[WS_PVAL + b * NCHUNK + chunk] = sV[0];
        ws[WS_PIDX + b * NCHUNK + chunk]           = sI[0];
    }
}

// ---------------------------------------------------------------------------
// Kernel 1b: fold NCHUNK partials per batch, derive crop origin + disc radius.
// ---------------------------------------------------------------------------
__global__ __launch_bounds__(64) void roi_finalize_kernel(
    const float* __restrict__ outer_wh,  // [B,2,H,W]
    float* __restrict__ out,             // hook coords go at the tail
    int* __restrict__ ws)
{
    const int b   = blockIdx.x;
    const int tid = threadIdx.x;

    __shared__ float sV[64];
    __shared__ int   sI[64];
    sV[tid] = ((const float*)ws)[WS_PVAL + b * NCHUNK + tid];
    sI[tid] = ws[WS_PIDX + b * NCHUNK + tid];
    __syncthreads();
    for (int s = 32; s > 0; s >>= 1) {
        if (tid < s) {
            const float ov = sV[tid + s]; const int oi = sI[tid + s];
            if (ov > sV[tid] || (ov == sV[tid] && oi < sI[tid])) { sV[tid] = ov; sI[tid] = oi; }
        }
        __syncthreads();
    }

    if (tid == 0) {
        const int p  = sI[0];
        const int cx = p / NW;   // row
        const int cy = p % NW;   // col
        int top  = cx - RADIUS; top  = top  < 0 ? 0 : (top  > NH - 1 - CROP ? NH - 1 - CROP : top);
        int left = cy - RADIUS; left = left < 0 ? 0 : (left > NW - 1 - CROP ? NW - 1 - CROP : left);
        const float r = outer_wh[((size_t)b * 2) * (size_t)(NH * NW) + (size_t)cx * NW + cy];
        int rad = (int)floorf(r + (float)R_EXPAND);
        rad = rad < 0 ? 0 : (rad > R_CLAMP ? R_CLAMP : rad);
        ws[WS_PARAM + b * 8 + 0] = cx;  ws[WS_PARAM + b * 8 + 1] = cy;
        ws[WS_PARAM + b * 8 + 2] = top; ws[WS_PARAM + b * 8 + 3] = left;
        ws[WS_PARAM + b * 8 + 4] = rad;
        float* hook = out + (size_t)NB * NC * CROP * CROP;
        hook[b * 4 + 0] = (float)top;
        hook[b * 4 + 1] = (float)(top + CROP);
        hook[b * 4 + 2] = (float)left;
        hook[b * 4 + 3] = (float)(left + CROP);
    }
}

// ---------------------------------------------------------------------------
// Kernel 2: one block per (b,c) plane. TDM DMA of the strided 128x128 f32
// tile (row stride 512) into LDS, then masked float4 writeback.
// ---------------------------------------------------------------------------
__global__ __launch_bounds__(256) void crop_mask_kernel(
    const float* __restrict__ feature_map,  // [B,C,H,W]
    const int* __restrict__ ws,
    float* __restrict__ out)                // [B,C,128,128]
{
    __shared__ float tile[CROP * CROP];     // 64 KB of the 320 KB WGP LDS

    const int bc  = blockIdx.x;
    const int b   = bc / NC;
    const int c   = bc % NC;
    const int tid = threadIdx.x;

    const int cx   = ws[WS_PARAM + b * 8 + 0];
    const int cy   = ws[WS_PARAM + b * 8 + 1];
    const int top  = ws[WS_PARAM + b * 8 + 2];
    const int left = ws[WS_PARAM + b * 8 + 3];
    const int rad  = ws[WS_PARAM + b * 8 + 4];

    const float* __restrict__ src =
        feature_map + (((size_t)b * NC + c) * (size_t)NH + (size_t)top) * NW + left;

#if defined(USE_TDM)
    if (tid < 32) {   // one wave issues the tensor DMA
        const u64 ga  = (u64)(uintptr_t)src;                 // tile start, 57-bit
        const u32 lds = (u32)(uintptr_t)&tile[0];            // wave-relative LDS byte addr
        u32x4 g0;
        g0.x = 1u;                                           // count=1, user descriptor
        g0.y = (u32)__builtin_amdgcn_readfirstlane((int)lds);
        g0.z = (u32)__builtin_amdgcn_readfirstlane((int)(u32)ga);
        g0.w = ((u32)__builtin_amdgcn_readfirstlane((int)(u32)(ga >> 32)) & 0x01FFFFFFu)
               | (2u << 30);                                 // type=2 ("image")
        i32x8 g1;
        g1[0] = (int)(2u << 16);            // data_size=4B, no multicast/iterate/pad
        g1[1] = (int)((u32)NW << 16);       // tensor_dim0 = 512 (lo16 in [63:48])
        g1[2] = (int)((u32)NH << 16);       // tensor_dim1 = 512 (lo16 in [95:80])
        g1[3] = (int)((u32)CROP << 16);     // tile_dim0 = 128   ([127:112])
        g1[4] = (int)CROP;                  // tile_dim1 = 128, tile_dim2 = 0
        g1[5] = (int)NW;                    // tensor_dim0_stride = 512 elements
        g1[6] = 0;                          // stride0 hi / stride1 lo
        g1[7] = 0;                          // stride1 hi
        const i32x4 gz4 = {0, 0, 0, 0};     // groups 2/3 unused (2D tile)
        const i32x8 gz8 = {0, 0, 0, 0, 0, 0, 0, 0};
        __builtin_amdgcn_tensor_load_to_lds(g0, g1, gz4, gz4, gz8, 0);
#if __has_builtin(__builtin_amdgcn_s_wait_tensorcnt)
        __builtin_amdgcn_s_wait_tensorcnt(0);
#else
        asm volatile("s_wait_tensorcnt 0x0" ::: "memory");
#endif
    }
#else
    // Fallback staging: per-lane b128 copies into LDS
    #pragma unroll
    for (int k = 0; k < 16; ++k) {
        const int p4 = tid + k * 256;
        const int i  = p4 >> 5;
        const int j  = (p4 & 31) << 2;
#if defined(USE_ASYNC_LDS)
        __builtin_amdgcn_global_load_async_to_lds_b128(
            (const __attribute__((address_space(1))) void*)(uintptr_t)(src + (size_t)i * NW + j),
            (__attribute__((address_space(3))) void*)(uintptr_t)&tile[p4 << 2], 0, 0);
#else
        *(float4*)&tile[p4 << 2] = *(const float4*)(src + (size_t)i * NW + j);
#endif
    }
#if defined(USE_ASYNC_LDS)
    __builtin_amdgcn_s_wait_asynccnt(0);
#endif
#endif
    __syncthreads();

    // Disc mask + writeback: 16 float4 per thread.
    const int rad2 = rad * rad;
    float4* __restrict__ outp = (float4*)(out + (size_t)bc * (CROP * CROP));
    const float4* __restrict__ tp = (const float4*)tile;
    #pragma unroll
    for (int k = 0; k < 16; ++k) {
        const int p4 = tid + k * 256;
        const int i  = p4 >> 5;             // crop row
        const int j  = (p4 & 31) << 2;      // crop col (x4)
        float4 v = tp[p4];
        const int dy  = top + i - cx;
        const int dy2 = dy * dy;
        const int dx0 = left + j - cy;
        if (dy2 + (dx0 + 0) * (dx0 + 0) > rad2) v.x = 0.0f;
        if (dy2 + (dx0 + 1) * (dx0 + 1) > rad2) v.y = 0.0f;
        if (dy2 + (dx0 + 2) * (dx0 + 2) > rad2) v.z = 0.0f;
        if (dy2 + (dx0 + 3) * (dx0 + 3) > rad2) v.w = 0.0f;
        outp[p4] = v;
    }
}

// ---------------------------------------------------------------------------
extern "C" void kernel_launch(void* const* d_in, const int* in_sizes, int n_in,
                              void* d_out, int out_size, void* d_ws, size_t ws_size,
                              hipStream_t stream) {
    const float* feature_map = (const float*)d_in[0];
    const float* heatmap     = (const float*)d_in[1];
    const float* outer_wh    = (const float*)d_in[2];
    float* out = (float*)d_out;
    int*   ws  = (int*)d_ws;

    nms_partial_kernel<<<NB * NCHUNK, 256, 0, stream>>>(heatmap, ws);
    roi_finalize_kernel<<<NB, 64, 0, stream>>>(outer_wh, out, ws);
    crop_mask_kernel<<<NB * NC, 256, 0, stream>>>(feature_map, ws, out);
}